// Transformer_28578712387756
// MI455X (gfx1250) — compile-verified
//
#include <hip/hip_runtime.h>
#include <math.h>

typedef __attribute__((ext_vector_type(2))) float v2f;
typedef __attribute__((ext_vector_type(8))) float v8f;

#define KNN_BLK 256
#define PT 8            // point-tiles (of 16) swept per wave in proj_kernel

// ---------------------------------------------------------------------------
// Projection GEMMs via V_WMMA_F32_16X16X4_F32.
// Each wave: one 16-row M-tile of W, swept across PT consecutive 16-point
// tiles. A fragments (weights) loaded once per wave into registers. Per tile:
// phase 1 loads all 16 B fragments (32 outstanding loads, one wait), phase 2
// runs 16 WMMAs back-to-back on two independent accumulators.
// Output stored point-major out[b][n][c].
// grid.x = B * N/(16*PT), grid.y = 0:q 1:k 2..5:v m-tiles. 32 thr = 1 wave.
// ---------------------------------------------------------------------------
__global__ __launch_bounds__(32) void proj_kernel(
    const float* __restrict__ x,                       // (B, Cin=64, N)
    const float* __restrict__ Wq, const float* __restrict__ bq,
    const float* __restrict__ Wk, const float* __restrict__ bk,
    const float* __restrict__ Wv, const float* __restrict__ bv,
    float* __restrict__ xq, float* __restrict__ xk, float* __restrict__ xv,
    int N)
{
    const int sweeps = N / (16 * PT);          // point sweeps per batch
    const int b     = blockIdx.x / sweeps;
    const int nbase = (blockIdx.x % sweeps) * (16 * PT);
    const int which = blockIdx.y;

    const float* W; const float* bias; float* out; int m0; int Cout;
    if (which == 0)      { W = Wq; bias = bq; out = xq; m0 = 0;                Cout = 16; }
    else if (which == 1) { W = Wk; bias = bk; out = xk; m0 = 0;                Cout = 16; }
    else                 { W = Wv; bias = bv; out = xv; m0 = (which - 2) * 16; Cout = 64; }

    const int lane = threadIdx.x;      // 0..31
    const int half = lane >> 4;        // selects K pair within each K4 step
    const int l15  = lane & 15;

    // ---- hoist A fragments: 16 K-steps x v2f, reused across all PT tiles ---
    // A (16x4): lane = row m; VGPR j holds K = 4*t + 2*half + j
    v2f a[16];
    {
        const float* wrow = W + (m0 + l15) * 64 + 2 * half;
        #pragma unroll
        for (int t = 0; t < 16; ++t) {
            a[t].x = wrow[4 * t + 0];
            a[t].y = wrow[4 * t + 1];
        }
    }
    // ---- hoist bias for this lane's 8 output channels ----------------------
    const int cbase = m0 + 8 * half;
    float bsl[8];
    #pragma unroll
    for (int r = 0; r < 8; ++r) bsl[r] = bias[cbase + r];

    const float* Xb = x + b * (64 * N);    // 32-bit offsets throughout
    float* outb = out + b * (N * Cout);

    for (int nt = 0; nt < PT; ++nt) {
        const int n0 = nbase + nt * 16;

        // ---- phase 1: stage all B fragments for this tile ------------------
        // B (4x16): lane = column n; VGPR j holds K = 4*t + 2*half + j
        v2f bfrag[16];
        #pragma unroll
        for (int t = 0; t < 16; ++t) {
            const float* xcol = Xb + (4 * t + 2 * half) * N + (n0 + l15);
            bfrag[t].x = xcol[0];
            bfrag[t].y = xcol[N];
        }

        // ---- phase 2: 16 WMMAs on two independent accumulator chains -------
        v8f acc0 = {};
        v8f acc1 = {};
        #pragma unroll
        for (int t = 0; t < 16; t += 2) {
            acc0 = __builtin_amdgcn_wmma_f32_16x16x4_f32(
                false, a[t],     false, bfrag[t],     (short)0, acc0, false, false);
            acc1 = __builtin_amdgcn_wmma_f32_16x16x4_f32(
                false, a[t + 1], false, bfrag[t + 1], (short)0, acc1, false, false);
        }

        // D layout: lane(0..15) = point column, VGPR r = channel r + 8*half.
        v8f vout;
        #pragma unroll
        for (int r = 0; r < 8; ++r) vout[r] = (acc0[r] + acc1[r]) + bsl[r];
        float* o = outb + (n0 + l15) * Cout + cbase;   // 32B-aligned
        *(v8f*)o = vout;
    }
}

// ---------------------------------------------------------------------------
// Brute-force kNN (k=16, includes self). One thread per query point;
// candidate coordinates staged through LDS in 256-point tiles.
// Top-16 kept as a fully unrolled sorted insertion list in registers.
// ---------------------------------------------------------------------------
__global__ __launch_bounds__(KNN_BLK) void knn_kernel(
    const float* __restrict__ p,   // (B, 3, N)
    int* __restrict__ nbr,         // (B, N, 16)
    int N)
{
    __shared__ float sx[KNN_BLK], sy[KNN_BLK], sz[KNN_BLK];

    const int tilesPerB = N / KNN_BLK;
    const int b  = blockIdx.x / tilesPerB;
    const int q0 = (blockIdx.x % tilesPerB) * KNN_BLK;

    const float* px = p + b * (3 * N);
    const float* py = px + N;
    const float* pz = py + N;

    const int qi = q0 + threadIdx.x;
    const float qx = px[qi], qy = py[qi], qz = pz[qi];

    float bd[16]; int bi[16];
    #pragma unroll
    for (int j = 0; j < 16; ++j) { bd[j] = 3.4e38f; bi[j] = 0; }

    for (int c0 = 0; c0 < N; c0 += KNN_BLK) {
        __syncthreads();
        sx[threadIdx.x] = px[c0 + threadIdx.x];
        sy[threadIdx.x] = py[c0 + threadIdx.x];
        sz[threadIdx.x] = pz[c0 + threadIdx.x];
        if (c0 + KNN_BLK < N) {  // speculative prefetch of next candidate tile
            __builtin_prefetch(px + c0 + KNN_BLK + threadIdx.x, 0, 0);
            __builtin_prefetch(py + c0 + KNN_BLK + threadIdx.x, 0, 0);
            __builtin_prefetch(pz + c0 + KNN_BLK + threadIdx.x, 0, 0);
        }
        __syncthreads();

        #pragma unroll 4
        for (int c = 0; c < KNN_BLK; ++c) {
            const float dx = sx[c] - qx;
            const float dy = sy[c] - qy;
            const float dz = sz[c] - qz;
            const float d = fmaf(dx, dx, fmaf(dy, dy, dz * dz));
            if (d < bd[15]) {
                const int ci = c0 + c;
                bool placed = false;
                #pragma unroll
                for (int j = 15; j > 0; --j) {
                    const bool shift = !placed && (d < bd[j - 1]);
                    if (shift) { bd[j] = bd[j - 1]; bi[j] = bi[j - 1]; }
                    else if (!placed) { bd[j] = d; bi[j] = ci; placed = true; }
                }
                if (!placed) { bd[0] = d; bi[0] = ci; }
            }
        }
    }

    int* o = nbr + (b * N + qi) * 16;
    #pragma unroll
    for (int j = 0; j < 16; ++j) o[j] = bi[j];
}

// ---------------------------------------------------------------------------
// Grouped vector attention with two-pass softmax, one thread per point.
// Pass 1: grouped logits -> per-group max.  Pass 2: recompute logits,
// accumulate exp-weighted values + denominator simultaneously, then scale.
// logit[g] = sum_{m in {g,g+8}} ( q[m]*k_j[m] + Wp[m]·rel + bp[m] )
// out[c]   = softmax_s(logit[c%8]) · v_j[c]
// ---------------------------------------------------------------------------
__global__ __launch_bounds__(256) void attn_kernel(
    const float* __restrict__ p,    // (B, 3, N)
    const float* __restrict__ xq,   // (B, N, 16)
    const float* __restrict__ xk,   // (B, N, 16)
    const float* __restrict__ xv,   // (B, N, 64)
    const int*   __restrict__ nbr,  // (B, N, 16)
    const float* __restrict__ Wp,   // (16, 3)
    const float* __restrict__ bp,   // (16)
    float* __restrict__ out,        // (B, 64, N)
    int N)
{
    __shared__ float sWp[48], sbp[16];
    if (threadIdx.x < 48) sWp[threadIdx.x] = Wp[threadIdx.x];
    if (threadIdx.x < 16) sbp[threadIdx.x] = bp[threadIdx.x];
    __syncthreads();

    const int gid = blockIdx.x * blockDim.x + threadIdx.x;
    const int b = gid / N;
    const int n = gid % N;

    const float* px = p + b * (3 * N);
    const float* py = px + N;
    const float* pz = py + N;
    const float cx = px[n], cy = py[n], cz = pz[n];

    float q[16];
    {
        const float* qp = xq + (b * N + n) * 16;
        #pragma unroll
        for (int m = 0; m < 16; ++m) q[m] = qp[m];
    }

    int idxs[16];
    {
        const int* ip = nbr + (b * N + n) * 16;
        #pragma unroll
        for (int s = 0; s < 16; ++s) idxs[s] = ip[s];
    }

    const float* xkb = xk + b * (N * 16);
    const float* xvb = xv + b * (N * 64);

    float gmax[8];
    #pragma unroll
    for (int g = 0; g < 8; ++g) gmax[g] = -3.4e38f;

    // ---- pass 1: per-group logit max ----
    for (int s = 0; s < 16; ++s) {
        const int j = idxs[s];
        const float rx = px[j] - cx, ry = py[j] - cy, rz = pz[j] - cz;
        const float* kp = xkb + j * 16;
        #pragma unroll
        for (int g = 0; g < 8; ++g) {
            float lg = q[g] * kp[g] + q[g + 8] * kp[g + 8];
            lg += fmaf(sWp[3*g + 0],      rx, fmaf(sWp[3*g + 1],      ry, sWp[3*g + 2]      * rz)) + sbp[g];
            lg += fmaf(sWp[3*(g+8) + 0],  rx, fmaf(sWp[3*(g+8) + 1],  ry, sWp[3*(g+8) + 2]  * rz)) + sbp[g + 8];
            gmax[g] = fmaxf(gmax[g], lg);
        }
    }

    // ---- pass 2: exp-weighted accumulation + denominator ----
    float gsum[8];
    float acc[64];
    #pragma unroll
    for (int g = 0; g < 8; ++g) gsum[g] = 0.0f;
    #pragma unroll
    for (int c = 0; c < 64; ++c) acc[c] = 0.0f;

    for (int s = 0; s < 16; ++s) {
        const int j = idxs[s];
        const float rx = px[j] - cx, ry = py[j] - cy, rz = pz[j] - cz;
        const float* kp = xkb + j * 16;
        float e[8];
        #pragma unroll
        for (int g = 0; g < 8; ++g) {
            float lg = q[g] * kp[g] + q[g + 8] * kp[g + 8];
            lg += fmaf(sWp[3*g + 0],      rx, fmaf(sWp[3*g + 1],      ry, sWp[3*g + 2]      * rz)) + sbp[g];
            lg += fmaf(sWp[3*(g+8) + 0],  rx, fmaf(sWp[3*(g+8) + 1],  ry, sWp[3*(g+8) + 2]  * rz)) + sbp[g + 8];
            e[g] = __expf(lg - gmax[g]);
            gsum[g] += e[g];
        }
        const float* vp = xvb + j * 64;
        #pragma unroll
        for (int c = 0; c < 64; ++c)
            acc[c] = fmaf(e[c & 7], vp[c], acc[c]);
    }

    float rs[8];
    #pragma unroll
    for (int g = 0; g < 8; ++g) rs[g] = 1.0f / gsum[g];

    float* o = out + b * (64 * N) + n;
    #pragma unroll
    for (int c = 0; c < 64; ++c)
        o[c * N] = acc[c] * rs[c & 7];
}

// ---------------------------------------------------------------------------
extern "C" void kernel_launch(void* const* d_in, const int* in_sizes, int n_in,
                              void* d_out, int out_size, void* d_ws, size_t ws_size,
                              hipStream_t stream) {
    const float* p  = (const float*)d_in[0];
    const float* x  = (const float*)d_in[1];
    const float* Wq = (const float*)d_in[2];
    const float* bq = (const float*)d_in[3];
    const float* Wk = (const float*)d_in[4];
    const float* bk = (const float*)d_in[5];
    const float* Wv = (const float*)d_in[6];
    const float* bv = (const float*)d_in[7];
    const float* Wp = (const float*)d_in[8];
    const float* bp = (const float*)d_in[9];
    float* out = (float*)d_out;

    const int B = 2, N = 8192;

    float* xq = (float*)d_ws;                      // B*N*16
    float* xk = xq + (size_t)B * N * 16;           // B*N*16
    float* xv = xk + (size_t)B * N * 16;           // B*N*64
    int*  nbr = (int*)(xv + (size_t)B * N * 64);   // B*N*16

    dim3 pgrid(B * (N / (16 * PT)), 6);
    proj_kernel<<<pgrid, 32, 0, stream>>>(x, Wq, bq, Wk, bk, Wv, bv, xq, xk, xv, N);

    knn_kernel<<<B * (N / KNN_BLK), KNN_BLK, 0, stream>>>(p, nbr, N);

    attn_kernel<<<(B * N) / 256, 256, 0, stream>>>(p, xq, xk, xv, nbr, Wp, bp, out, N);
}